// AttentionBlock_21036749816213
// MI455X (gfx1250) — compile-verified
//
#include <hip/hip_runtime.h>
#include <math.h>

#define B_     4
#define L_     4096
#define INDIM  512
#define D_     64
#define OUTDIM 512
#define KT     128   // K-tile staged in LDS for the AV kernel (128x64 f32 = 32 KB)

typedef __attribute__((ext_vector_type(2))) float v2f;
typedef __attribute__((ext_vector_type(8))) float v8f;

// D = A(16x4 f32) * B(4x16 f32) + C(16x16 f32), full wave32 op.
__device__ __forceinline__ v8f wmma_f32(v2f a, v2f b, v8f c) {
    return __builtin_amdgcn_wmma_f32_16x16x4_f32(
        /*neg_a=*/false, a, /*neg_b=*/false, b,
        /*c_mod=*/(short)0, c, /*reuse_a=*/false, /*reuse_b=*/false);
}

// ---------------- K0: zero accumulators (colsum, AV) ----------------------
__global__ void zero_kernel(float* __restrict__ p, int n) {
    int i = blockIdx.x * blockDim.x + threadIdx.x;
    if (i < n) p[i] = 0.0f;
}

// ---------------- K1: QKV projection  [16384,512] @ [512,64] + bias -------
// blockIdx.x = M-tile (1024), threadIdx.y = N-tile (4), blockIdx.y = q/k/v
__global__ void qkv_kernel(const float* __restrict__ x,
                           const float* __restrict__ Wq, const float* __restrict__ bq,
                           const float* __restrict__ Wk, const float* __restrict__ bk,
                           const float* __restrict__ Wv, const float* __restrict__ bv,
                           float* __restrict__ Q, float* __restrict__ Kmat,
                           float* __restrict__ V) {
    const int lane = threadIdx.x;
    const int lm = lane & 15, hi = lane >> 4;
    const int m0 = blockIdx.x * 16;
    const int n0 = threadIdx.y * 16;
    const int which = blockIdx.y;

    const float* W; const float* bias; float* Out;
    if (which == 0)      { W = Wq; bias = bq; Out = Q; }
    else if (which == 1) { W = Wk; bias = bk; Out = Kmat; }
    else                 { W = Wv; bias = bv; Out = V; }

    const float* xrow = x + (size_t)(m0 + lm) * INDIM;
    v8f c = {};
    for (int k = 0; k < INDIM; k += 4) {
        v2f a, bf;
        a.x  = xrow[k + 2 * hi];
        a.y  = xrow[k + 2 * hi + 1];
        bf.x = W[(size_t)(k + 2 * hi)     * D_ + n0 + lm];
        bf.y = W[(size_t)(k + 2 * hi + 1) * D_ + n0 + lm];
        c = wmma_f32(a, bf, c);
    }
    const float bb = bias[n0 + lm];
#pragma unroll
    for (int i = 0; i < 8; ++i)
        Out[(size_t)(m0 + hi * 8 + i) * D_ + n0 + lm] = c[i] + bb;
}

// ---------------- K2: scores = Q*K^T/8, write exp(), column sums ----------
// blockIdx.x = k-tile (256), blockIdx.y*4+threadIdx.y = q-tile (256), z = b
__global__ void scores_kernel(const float* __restrict__ Q,
                              const float* __restrict__ Kmat,
                              float* __restrict__ attn,
                              float* __restrict__ colsum) {
    const int lane = threadIdx.x;
    const int lm = lane & 15, hi = lane >> 4;
    const int k0 = blockIdx.x * 16;
    const int q0 = (blockIdx.y * blockDim.y + threadIdx.y) * 16;
    const int b  = blockIdx.z;

    const float* qrow = Q    + (size_t)(b * L_ + q0 + lm) * D_;
    const float* krow = Kmat + (size_t)(b * L_ + k0 + lm) * D_;  // B = K^T

    v8f c = {};
#pragma unroll
    for (int kd = 0; kd < D_; kd += 4) {
        v2f a, bf;
        a.x  = qrow[kd + 2 * hi];
        a.y  = qrow[kd + 2 * hi + 1];
        bf.x = krow[kd + 2 * hi];
        bf.y = krow[kd + 2 * hi + 1];
        c = wmma_f32(a, bf, c);
    }
    const float scale = 0.125f;  // 1/sqrt(64)
    float part = 0.0f;
#pragma unroll
    for (int i = 0; i < 8; ++i) {
        float e = __expf(c[i] * scale);  // scores ~ N(0,1): safe w/o max-shift
        attn[((size_t)b * L_ + q0 + hi * 8 + i) * L_ + k0 + lm] = e;
        part += e;
    }
    // combine the two row-halves (M 0-7 in lanes 0-15, M 8-15 in lanes 16-31)
    part += __shfl_xor(part, 16, 32);
    if (lane < 16)
        atomicAdd(&colsum[b * L_ + k0 + lane], part);  // softmax over axis=1
}

// ---------------- K3a: colsum -> 1/colsum (in place, 16K elements) --------
__global__ void rcp_kernel(float* __restrict__ p, int n) {
    int i = blockIdx.x * blockDim.x + threadIdx.x;
    if (i < n) p[i] = 1.0f / p[i];
}

// ---------------- K3b: normalize attention columns (streaming multiply) ---
__global__ void norm_kernel(float* __restrict__ attn,
                            const float* __restrict__ colinv) {
    const size_t total  = (size_t)B_ * L_ * L_;
    const size_t stride = (size_t)gridDim.x * blockDim.x * 4;
    for (size_t idx = ((size_t)blockIdx.x * blockDim.x + threadIdx.x) * 4;
         idx < total; idx += stride) {
        float4 v = *(const float4*)(attn + idx);
        const int b = (int)(idx / ((size_t)L_ * L_));
        const int k = (int)(idx % L_);           // L_ divisible by 4
        const float* cs = colinv + b * L_ + k;
        v.x *= cs[0]; v.y *= cs[1]; v.z *= cs[2]; v.w *= cs[3];
        *(float4*)(attn + idx) = v;
    }
}

// ---------------- K4: AV = attn(4096x4096) @ V(4096x64) per batch ---------
// One wave computes a 16(q) x 64(d) output block (4 accumulators): each
// attention fragment load feeds 4 WMMAs. V is staged 128x64 in LDS per block
// and served to all 4 waves from ds_load. K range split 4-ways over
// blockIdx.z, accumulated into AV with f32 global atomics.
// grid: (64 q-groups, B, 4 k-chunks), block (32,4): threadIdx.y = q-subtile.
__global__ void av_kernel(const float* __restrict__ attn,
                          const float* __restrict__ V,
                          float* __restrict__ AV) {
    __shared__ float vtile[KT * D_];   // 32 KB
    const int lane = threadIdx.x;
    const int lm = lane & 15, hi = lane >> 4;
    const int q0 = (blockIdx.x * blockDim.y + threadIdx.y) * 16;
    const int b  = blockIdx.y;
    const int kchunk0 = blockIdx.z * (L_ / 4);   // 1024-wide K slice
    const int tid = threadIdx.y * 32 + lane;     // 0..127

    const float* arow  = attn + ((size_t)b * L_ + q0 + lm) * L_ + kchunk0;
    const float* vbase = V + ((size_t)b * L_ + kchunk0) * D_;

    v8f c0 = {}, c1 = {}, c2 = {}, c3 = {};

    for (int kb = 0; kb < L_ / 4; kb += KT) {
        // cooperative stage: 128 rows x 64 cols = 8192 f32, 128 thr x float4
        const float4* src = (const float4*)(vbase + (size_t)kb * D_);
        float4*       dst = (float4*)vtile;
#pragma unroll
        for (int i = 0; i < (KT * D_ / 4) / 128; ++i)
            dst[tid + i * 128] = src[tid + i * 128];
        __syncthreads();

#pragma unroll 4
        for (int kk = 0; kk < KT; kk += 4) {
            v2f a;
            a.x = arow[kb + kk + 2 * hi];
            a.y = arow[kb + kk + 2 * hi + 1];
            const float* l0 = vtile + (kk + 2 * hi) * D_;
            const float* l1 = l0 + D_;
            v2f bf;
            bf.x = l0[lm];      bf.y = l1[lm];      c0 = wmma_f32(a, bf, c0);
            bf.x = l0[16 + lm]; bf.y = l1[16 + lm]; c1 = wmma_f32(a, bf, c1);
            bf.x = l0[32 + lm]; bf.y = l1[32 + lm]; c2 = wmma_f32(a, bf, c2);
            bf.x = l0[48 + lm]; bf.y = l1[48 + lm]; c3 = wmma_f32(a, bf, c3);
        }
        __syncthreads();
    }

#pragma unroll
    for (int i = 0; i < 8; ++i) {
        float* o = AV + ((size_t)b * L_ + q0 + hi * 8 + i) * D_;
        atomicAdd(o + lm,      c0[i]);
        atomicAdd(o + 16 + lm, c1[i]);
        atomicAdd(o + 32 + lm, c2[i]);
        atomicAdd(o + 48 + lm, c3[i]);
    }
}

// ---------------- K5: output = AV(16384x64) @ Wo(64x512) + bo -------------
// blockIdx.x = M-tile (1024), blockIdx.y*4+threadIdx.y = N-tile (32)
__global__ void outproj_kernel(const float* __restrict__ AV,
                               const float* __restrict__ Wo,
                               const float* __restrict__ bo,
                               float* __restrict__ out) {
    const int lane = threadIdx.x;
    const int lm = lane & 15, hi = lane >> 4;
    const int m0 = blockIdx.x * 16;
    const int n0 = (blockIdx.y * blockDim.y + threadIdx.y) * 16;

    const float* arow = AV + (size_t)(m0 + lm) * D_;
    v8f c = {};
#pragma unroll
    for (int k = 0; k < D_; k += 4) {
        v2f a, bf;
        a.x  = arow[k + 2 * hi];
        a.y  = arow[k + 2 * hi + 1];
        bf.x = Wo[(size_t)(k + 2 * hi)     * OUTDIM + n0 + lm];
        bf.y = Wo[(size_t)(k + 2 * hi + 1) * OUTDIM + n0 + lm];
        c = wmma_f32(a, bf, c);
    }
    const float bb = bo[n0 + lm];
#pragma unroll
    for (int i = 0; i < 8; ++i)
        out[(size_t)(m0 + hi * 8 + i) * OUTDIM + n0 + lm] = c[i] + bb;
}

extern "C" void kernel_launch(void* const* d_in, const int* in_sizes, int n_in,
                              void* d_out, int out_size, void* d_ws, size_t ws_size,
                              hipStream_t stream) {
    (void)in_sizes; (void)n_in; (void)out_size; (void)ws_size;

    const float* x  = (const float*)d_in[0];
    const float* Wq = (const float*)d_in[1];
    const float* bq = (const float*)d_in[2];
    const float* Wk = (const float*)d_in[3];
    const float* bk = (const float*)d_in[4];
    const float* Wv = (const float*)d_in[5];
    const float* bv = (const float*)d_in[6];
    const float* Wo = (const float*)d_in[7];
    const float* bo = (const float*)d_in[8];

    float* out  = (float*)d_out;                          // [B,L,OUTDIM]
    float* attn = out + (size_t)B_ * L_ * OUTDIM;         // [B,L,L]

    // workspace: Q,K,V,AV = 4 x 4MB, colsum = 64KB  (~16.8 MB total)
    float* Q      = (float*)d_ws;
    float* Kmat   = Q    + (size_t)B_ * L_ * D_;
    float* V      = Kmat + (size_t)B_ * L_ * D_;
    float* AV     = V    + (size_t)B_ * L_ * D_;
    float* colsum = AV   + (size_t)B_ * L_ * D_;

    const int nQKV = B_ * L_ * D_;   // 1,048,576

    zero_kernel<<<dim3((B_ * L_ + 255) / 256), dim3(256), 0, stream>>>(colsum, B_ * L_);
    zero_kernel<<<dim3((nQKV + 255) / 256), dim3(256), 0, stream>>>(AV, nQKV);

    qkv_kernel<<<dim3(B_ * L_ / 16, 3), dim3(32, 4), 0, stream>>>(
        x, Wq, bq, Wk, bk, Wv, bv, Q, Kmat, V);

    scores_kernel<<<dim3(L_ / 16, L_ / 64, B_), dim3(32, 4), 0, stream>>>(
        Q, Kmat, attn, colsum);

    rcp_kernel<<<dim3((B_ * L_ + 255) / 256), dim3(256), 0, stream>>>(colsum, B_ * L_);

    norm_kernel<<<dim3(8192), dim3(256), 0, stream>>>(attn, colsum);

    av_kernel<<<dim3(L_ / 64, B_, 4), dim3(32, 4), 0, stream>>>(attn, V, AV);

    outproj_kernel<<<dim3(B_ * L_ / 16, OUTDIM / 64), dim3(32, 4), 0, stream>>>(
        AV, Wo, bo, out);
}